// LinkPredict_13511967113605
// MI455X (gfx1250) — compile-verified
//
#include <hip/hip_runtime.h>

// Problem constants (match reference setup_inputs)
#define NNODES   50000
#define HDIM     500
#define NBASES   100
#define SUBD     5
#define NEDGES   100000

typedef __attribute__((ext_vector_type(2))) float v2f;
typedef __attribute__((ext_vector_type(8))) float v8f;

// ---------------------------------------------------------------------------
// Grid-stride zero fill (graph-capture safe; no hipMemset)
// ---------------------------------------------------------------------------
__global__ void zero_kernel(float* __restrict__ p, long n) {
    long i = (long)blockIdx.x * blockDim.x + threadIdx.x;
    long stride = (long)gridDim.x * blockDim.x;
    for (; i < n; i += stride) p[i] = 0.0f;
}

// ---------------------------------------------------------------------------
// Pack loop_w [K, N] into K-pair-interleaved layout so a B fragment
// (two consecutive K rows at one column) is a single aligned 8-byte load:
//   Wp[(kp*N + c)*2 + 0] = Wl[2*kp  ][c]
//   Wp[(kp*N + c)*2 + 1] = Wl[2*kp+1][c]
// ---------------------------------------------------------------------------
__global__ void pack_w_kernel(const float* __restrict__ Wl,
                              float* __restrict__ Wp) {
    const int K = HDIM, N = HDIM;
    int idx = blockIdx.x * blockDim.x + threadIdx.x;   // over (K/2)*N
    int total = (K >> 1) * N;
    if (idx >= total) return;
    int kp = idx / N;
    int c  = idx - kp * N;
    Wp[(long)idx * 2 + 0] = Wl[(long)(2 * kp)     * N + c];
    Wp[(long)idx * 2 + 1] = Wl[(long)(2 * kp + 1) * N + c];
}

// ---------------------------------------------------------------------------
// BDD edge-message kernel: one block per edge, thread b = basis block.
// msg[b*5+j] = sum_i x[src][b*5+i] * W[etype][b][i][j] ; * norm ; atomic into agg[dst]
// Scatter uses relaxed agent-scope fp32 atomics -> global_atomic_add_f32.
// Weight tables (4.74 MB) and feature matrix (100 MB) are L2-resident (192 MB).
// ---------------------------------------------------------------------------
__global__ void bdd_edge_kernel(const float* __restrict__ x,
                                const int*   __restrict__ rowmap,   // nullable
                                const int*   __restrict__ src,
                                const int*   __restrict__ dst,
                                const int*   __restrict__ etype,
                                const float* __restrict__ norm,
                                const float* __restrict__ W,        // [R, B, 5, 5]
                                float*       __restrict__ agg,      // [N, 500]
                                int numEdges) {
    int e = blockIdx.x;
    if (e >= numEdges) return;
    int b = threadIdx.x;
    if (b >= NBASES) return;

    int s = src[e];
    if (rowmap) s = rowmap[s];
    int d = dst[e];
    int t = etype[e];
    float nrm = norm[e];

    const float* xe = x + (long)s * HDIM + b * SUBD;
    const float* we = W + ((long)t * NBASES + b) * (SUBD * SUBD);

    float xi0 = xe[0], xi1 = xe[1], xi2 = xe[2], xi3 = xe[3], xi4 = xe[4];

    float* out = agg + (long)d * HDIM + b * SUBD;
#pragma unroll
    for (int j = 0; j < SUBD; ++j) {
        float acc = xi0 * we[0 * SUBD + j]
                  + xi1 * we[1 * SUBD + j]
                  + xi2 * we[2 * SUBD + j]
                  + xi3 * we[3 * SUBD + j]
                  + xi4 * we[4 * SUBD + j];
        (void)__hip_atomic_fetch_add(&out[j], acc * nrm,
                                     __ATOMIC_RELAXED, __HIP_MEMORY_SCOPE_AGENT);
    }
}

// ---------------------------------------------------------------------------
// Fused dense kernel: out = [relu]( agg + X @ Wl + bias ), Wl pre-packed.
// One wave per 16x64 tile, 4 f32 accumulators (V_WMMA_F32_16X16X4_F32).
// K = 500 split into 25 groups of 5 K-steps; two full fragment buffers are
// double-buffered at GROUP granularity with a manual 2x-unrolled loop, so both
// buffers are live simultaneously -> the register allocator cannot coalesce
// them -> 25 b64 loads stay in flight across 20 WMMAs (one wait per group
// instead of one wait per WMMA).
// Fragment layouts per CDNA5 ISA 7.12.2:
//   A 16x4 : lanes 0-15 rows M, VGPR0/1 = K {0,1} (half 0) / {2,3} (half 1)
//   B 4x16 : lanes 0-15 cols N, packed pair row index = K-step + half
//   C 16x16: VGPR r -> row r (half 0) / r+8 (half 1), col = lane&15
// Tails: clamped load addresses + guarded stores only (EXEC all-1s at WMMAs).
// ---------------------------------------------------------------------------
#define NT_PER_WAVE 4                    // 4 x 16 = 64 output columns per wave
#define KSTEPS      (HDIM / 4)           // 125
#define GRP         5                    // K-steps per pipeline group
#define NGRP        (KSTEPS / GRP)       // 25

__device__ __forceinline__
void load_group(const v2f* __restrict__ aPtr,
                const v2f* const bp[NT_PER_WAVE],
                int grp, v2f a[GRP], v2f b[GRP][NT_PER_WAVE]) {
#pragma unroll
    for (int s = 0; s < GRP; ++s) {
        int it = grp * GRP + s;
        a[s] = aPtr[2 * it];                       // b64, immediate offsets in-group
#pragma unroll
        for (int g = 0; g < NT_PER_WAVE; ++g)
            b[s][g] = bp[g][(long)it * HDIM];      // b64, offsets 0/4000/.../16000
    }
}

__device__ __forceinline__
void compute_group(const v2f a[GRP], const v2f b[GRP][NT_PER_WAVE],
                   v8f acc[NT_PER_WAVE]) {
#pragma unroll
    for (int s = 0; s < GRP; ++s)
#pragma unroll
        for (int g = 0; g < NT_PER_WAVE; ++g)
            // 8 args: (neg_a, A, neg_b, B, c_mod, C, reuse_a, reuse_b)
            acc[g] = __builtin_amdgcn_wmma_f32_16x16x4_f32(false, a[s], false, b[s][g],
                                                           (short)0, acc[g],
                                                           false, false);
}

__global__ void gemm_wmma_f32_kernel(const float* __restrict__ X,
                                     const int*   __restrict__ rowmap,   // nullable
                                     const float* __restrict__ Wp,       // packed [K/2, N] of v2f
                                     const float* __restrict__ bias,     // [N]
                                     const float* __restrict__ agg,      // [M, N]
                                     float*       __restrict__ out,      // [M, N]
                                     int M, int doRelu) {
    const int N = HDIM, K = HDIM;
    const int nBase = blockIdx.x * (16 * NT_PER_WAVE);
    const int mTile = blockIdx.y;
    const int lane  = threadIdx.x;      // 0..31, wave32
    const int half  = lane >> 4;        // 0 or 1
    const int l     = lane & 15;

    // Clamp OOB lanes to valid addresses (keeps EXEC all-1s for WMMA);
    // garbage values are never stored.
    int mRow = mTile * 16 + l;
    int mC   = mRow < M ? mRow : M - 1;
    int aRow = rowmap ? rowmap[mC] : mC;

    const v2f* aPtr = (const v2f*)(X + (long)aRow * K) + half;  // +2 per K-step

    const v2f* bp[NT_PER_WAVE];
    int col[NT_PER_WAVE];
#pragma unroll
    for (int g = 0; g < NT_PER_WAVE; ++g) {
        int c  = nBase + g * 16 + l;
        col[g] = c;
        int cc = c < N ? c : N - 1;
        bp[g]  = (const v2f*)Wp + (long)half * N + cc;          // +N per K-step
    }

    v8f acc[NT_PER_WAVE];
#pragma unroll
    for (int g = 0; g < NT_PER_WAVE; ++g)
        acc[g] = (v8f){0.f, 0.f, 0.f, 0.f, 0.f, 0.f, 0.f, 0.f};

    // ---- Group-level double-buffered pipeline (25 groups of 5 K-steps) ----
    v2f aA[GRP], bA[GRP][NT_PER_WAVE];
    v2f aB[GRP], bB[GRP][NT_PER_WAVE];

    load_group(aPtr, bp, 0, aA, bA);
    // grp = 1,3,...,23 : each body loads 2 groups, computes 2 groups
    for (int grp = 1; grp + 1 < NGRP; grp += 2) {
        load_group(aPtr, bp, grp, aB, bB);       // fill idle buffer
        compute_group(aA, bA, acc);              // consume live buffer (grp-1)
        load_group(aPtr, bp, grp + 1, aA, bA);
        compute_group(aB, bB, acc);              // consume (grp)
    }
    compute_group(aA, bA, acc);                  // last group (NGRP-1)

    // Epilogue: + agg + bias, optional ReLU, bounds-checked store.
#pragma unroll
    for (int g = 0; g < NT_PER_WAVE; ++g) {
        if (col[g] < N) {
            float bv = bias[col[g]];
#pragma unroll
            for (int r = 0; r < 8; ++r) {
                int row = mTile * 16 + r + 8 * half;
                if (row < M) {
                    long idx = (long)row * N + col[g];
                    float v = acc[g][r] + agg[idx] + bv;
                    if (doRelu) v = v > 0.f ? v : 0.f;
                    out[idx] = v;
                }
            }
        }
    }
}

// ---------------------------------------------------------------------------
// Launch:  layer1 = relu(agg1 + x @ lw1 + b1),  layer2 = agg2 + h1 @ lw2 + b2
// Workspace: agg [M*N] + h1 [M*N] + Wp1 [K*N] + Wp2 [K*N]  (~202 MB)
// ---------------------------------------------------------------------------
extern "C" void kernel_launch(void* const* d_in, const int* in_sizes, int n_in,
                              void* d_out, int out_size, void* d_ws, size_t ws_size,
                              hipStream_t stream) {
    const int*   nids  = (const int*)  d_in[0];
    const int*   src   = (const int*)  d_in[1];
    const int*   dst   = (const int*)  d_in[2];
    const int*   etype = (const int*)  d_in[3];
    const float* norm  = (const float*)d_in[4];
    const float* emb   = (const float*)d_in[5];
    const float* W1    = (const float*)d_in[6];
    const float* lw1   = (const float*)d_in[7];
    const float* b1    = (const float*)d_in[8];
    const float* W2    = (const float*)d_in[9];
    const float* lw2   = (const float*)d_in[10];
    const float* b2    = (const float*)d_in[11];
    float* out = (float*)d_out;

    const int M = in_sizes[0];      // 50000 nodes
    const int E = in_sizes[1];      // 100000 edges
    const int N = HDIM, K = HDIM;

    const long MN = (long)M * N;
    const long KN = (long)K * N;
    float* agg = (float*)d_ws;
    float* h1  = agg + MN;
    float* Wp1 = h1 + MN;
    float* Wp2 = Wp1 + KN;

    dim3 zGrid(2048), zBlk(256);
    dim3 eGrid((unsigned)E), eBlk(128);                 // 128 thr, b<100 active
    const int tilesN = (N + 16 * NT_PER_WAVE - 1) / (16 * NT_PER_WAVE);  // 8
    dim3 gGrid(tilesN, (M + 15) / 16), gBlk(32);        // one wave per 16x64 tile
    const int packTotal = (K / 2) * N;
    dim3 pGrid((packTotal + 255) / 256), pBlk(256);

    // ---- Pack loop weights (1 MB each; one-time per call) ----
    pack_w_kernel<<<pGrid, pBlk, 0, stream>>>(lw1, Wp1);
    pack_w_kernel<<<pGrid, pBlk, 0, stream>>>(lw2, Wp2);

    // ---- Layer 1 ----
    zero_kernel<<<zGrid, zBlk, 0, stream>>>(agg, MN);
    bdd_edge_kernel<<<eGrid, eBlk, 0, stream>>>(emb, nids, src, dst, etype, norm,
                                                W1, agg, E);
    gemm_wmma_f32_kernel<<<gGrid, gBlk, 0, stream>>>(emb, nids, Wp1, b1, agg, h1,
                                                     M, /*relu=*/1);

    // ---- Layer 2 ----
    zero_kernel<<<zGrid, zBlk, 0, stream>>>(agg, MN);
    bdd_edge_kernel<<<eGrid, eBlk, 0, stream>>>(h1, nullptr, src, dst, etype, norm,
                                                W2, agg, E);
    gemm_wmma_f32_kernel<<<gGrid, gBlk, 0, stream>>>(h1, nullptr, Wp2, b2, agg, out,
                                                     M, /*relu=*/0);
}